// GATLayer_20349555048923
// MI455X (gfx1250) — compile-verified
//
#include <hip/hip_runtime.h>
#include <math.h>

#define ALPHA 0.2f
#define Bsz 64
#define Nn  2048
#define Dd  1024
#define CH  16          // chunks over n per batch
#define NPC (Nn / CH)   // 128 rows per chunk
#define NPW (NPC / 8)   // 16 rows per wave

typedef float v2f __attribute__((ext_vector_type(2)));
typedef float v8f __attribute__((ext_vector_type(8)));

// workspace layout (float offsets)
#define WS_V1 0
#define WS_V2 1024
#define WS_CC 2048
#define WS_S0 2064
#define WS_PM 2176
#define WS_PS (WS_PM + Bsz * CH)
#define WS_PG (WS_PS + Bsz * CH)
#define WS_G  (WS_PG + Bsz * CH * Dd)

// v1[i] = sum_d W[d,i]*a_w[d]; v2[i] = sum_d W[d,i]*a_w[D+d]
__global__ void prep_v(const float* __restrict__ W, const float* __restrict__ aw,
                       float* __restrict__ ws) {
    int i = blockIdx.x * 256 + threadIdx.x;   // 0..1023
    float acc1 = 0.f, acc2 = 0.f;
    for (int d = 0; d < Dd; ++d) {
        float w = W[(size_t)d * Dd + i];
        acc1 = fmaf(w, aw[d], acc1);
        acc2 = fmaf(w, aw[Dd + d], acc2);
    }
    ws[WS_V1 + i] = acc1;
    ws[WS_V2 + i] = acc2;
}

// cc = w_b·a1 + w_b·a2 + a_b
__global__ void prep_cc(const float* __restrict__ wb, const float* __restrict__ aw,
                        const float* __restrict__ ab, float* __restrict__ ws) {
    __shared__ float red[256];
    int t = threadIdx.x;
    float p = 0.f;
    for (int d = t; d < Dd; d += 256) p = fmaf(wb[d], aw[d] + aw[Dd + d], p);
    red[t] = p; __syncthreads();
    for (int s = 128; s > 0; s >>= 1) { if (t < s) red[t] += red[t + s]; __syncthreads(); }
    if (t == 0) ws[WS_CC] = red[0] + ab[0];
}

// s0[b] = feat[b,0,:]·v1 + cc   (per-batch constant added to every logit)
__global__ void s0_kernel(const float* __restrict__ feat, float* __restrict__ ws) {
    int b = blockIdx.x, t = threadIdx.x;
    const float4* f  = (const float4*)(feat + (size_t)b * Nn * Dd);
    const float4* v1 = (const float4*)(ws + WS_V1);
    float4 a = f[t], v = v1[t];
    float p = a.x * v.x + a.y * v.y + a.z * v.z + a.w * v.w;
    __shared__ float red[256];
    red[t] = p; __syncthreads();
    for (int s = 128; s > 0; s >>= 1) { if (t < s) red[t] += red[t + s]; __syncthreads(); }
    if (t == 0) ws[WS_S0 + b] = red[0] + ws[WS_CC];
}

// Streaming pass, barrier-free inner loop: each wave owns 16 full rows,
// computes the 1024-wide logit dot with shfl_xor allreduce (wave32), and
// maintains private online-softmax state (m, s, G). One LDS merge per block.
__global__ void __launch_bounds__(256) flash_kernel(const float* __restrict__ feat,
                                                    float* __restrict__ ws) {
    const int c = blockIdx.x, b = blockIdx.y;
    const int t = threadIdx.x, lane = t & 31, wv = t >> 5;

    __shared__ float lG[8 * Dd];      // 32 KB: per-wave G partials
    __shared__ float lm[8], lsum[8];

    // v2 slices this lane needs: element j*128 + lane*4
    float4 v2r[8];
    #pragma unroll
    for (int j = 0; j < 8; ++j)
        v2r[j] = *(const float4*)(ws + WS_V2 + j * 128 + lane * 4);

    const float s0b = ws[WS_S0 + b];
    float4 G[8];
    #pragma unroll
    for (int j = 0; j < 8; ++j) G[j] = (float4){0.f, 0.f, 0.f, 0.f};
    float m = -3.4e38f, s = 0.f;

    const float* rowbase =
        feat + ((size_t)b * Nn + (size_t)c * NPC + (size_t)wv * NPW) * Dd;

    for (int n = 0; n < NPW; ++n) {
        const float* row = rowbase + (size_t)n * Dd;
        float4 f[8];
        #pragma unroll
        for (int j = 0; j < 8; ++j)
            f[j] = *(const float4*)(row + j * 128 + lane * 4);

        float p = 0.f;
        #pragma unroll
        for (int j = 0; j < 8; ++j)
            p += f[j].x * v2r[j].x + f[j].y * v2r[j].y +
                 f[j].z * v2r[j].z + f[j].w * v2r[j].w;
        #pragma unroll
        for (int off = 16; off > 0; off >>= 1)
            p += __shfl_xor(p, off, 32);          // allreduce within wave32

        float e = p + s0b;
        e = (e >= 0.f) ? e : ALPHA * e;           // leaky relu

        float mn = fmaxf(m, e);
        float sc = __expf(m - mn);                // 0 on first iter
        float w  = __expf(e - mn);
        s = s * sc + w;
        #pragma unroll
        for (int j = 0; j < 8; ++j) {
            G[j].x = G[j].x * sc + f[j].x * w;
            G[j].y = G[j].y * sc + f[j].y * w;
            G[j].z = G[j].z * sc + f[j].z * w;
            G[j].w = G[j].w * sc + f[j].w * w;
        }
        m = mn;
    }

    // single block-level merge of the 8 wave partials
    if (lane == 0) { lm[wv] = m; lsum[wv] = s; }
    #pragma unroll
    for (int j = 0; j < 8; ++j)
        *(float4*)(&lG[wv * Dd + j * 128 + lane * 4]) = G[j];
    __syncthreads();

    float M = -3.4e38f;
    #pragma unroll
    for (int w = 0; w < 8; ++w) M = fmaxf(M, lm[w]);
    float wsc[8], S = 0.f;
    #pragma unroll
    for (int w = 0; w < 8; ++w) {
        wsc[w] = __expf(lm[w] - M);
        S += lsum[w] * wsc[w];
    }

    // thread t owns output elements t*4 .. t*4+3
    float4 acc = {0.f, 0.f, 0.f, 0.f};
    #pragma unroll
    for (int w = 0; w < 8; ++w) {
        float4 gv = *(const float4*)(&lG[w * Dd + t * 4]);
        float x = wsc[w];
        acc.x += gv.x * x; acc.y += gv.y * x; acc.z += gv.z * x; acc.w += gv.w * x;
    }

    int pi = b * CH + c;
    if (t == 0) { ws[WS_PM + pi] = M; ws[WS_PS + pi] = S; }
    ((float4*)(ws + WS_PG + (size_t)pi * Dd))[t] = acc;
}

// merge chunk partials -> g[b, :] (attention-weighted feature pool)
__global__ void combine_kernel(float* __restrict__ ws) {
    int b = blockIdx.x, t = threadIdx.x;
    __shared__ float wgt[CH];
    if (t == 0) {
        float M = -3.4e38f;
        for (int c = 0; c < CH; ++c) M = fmaxf(M, ws[WS_PM + b * CH + c]);
        float S = 0.f;
        for (int c = 0; c < CH; ++c) {
            float w = __expf(ws[WS_PM + b * CH + c] - M);
            wgt[c] = w;
            S += ws[WS_PS + b * CH + c] * w;
        }
        float inv = 1.f / S;
        for (int c = 0; c < CH; ++c) wgt[c] *= inv;
    }
    __syncthreads();
    float4 acc = {0.f, 0.f, 0.f, 0.f};
    for (int c = 0; c < CH; ++c) {
        float4 gv = ((const float4*)(ws + WS_PG + (size_t)(b * CH + c) * Dd))[t];
        float w = wgt[c];
        acc.x += gv.x * w; acc.y += gv.y * w; acc.z += gv.z * w; acc.w += gv.w * w;
    }
    ((float4*)(ws + WS_G + (size_t)b * Dd))[t] = acc;
}

// out[b,d] = relu( g[b,:]·W[d,:] + w_b[d] + bias[d] ) via V_WMMA_F32_16X16X4_F32.
// Block = 256 thr (8 waves), covers M=64 (all b) x N=32 (d). Grid = 1024/32.
__global__ void __launch_bounds__(256) wmma_gemm(const float* __restrict__ Wm,
                                                 const float* __restrict__ wb,
                                                 const float* __restrict__ bias,
                                                 float* __restrict__ out,
                                                 const float* __restrict__ ws) {
    const int nbase = blockIdx.x * 32;
    const int t = threadIdx.x, lane = t & 31, wv = t >> 5;
    const int mt = wv & 3, nt = wv >> 2;          // 4 M-tiles x 2 N-tiles
    __shared__ float Al[64 * 33];                 // A[m][k], padded
    __shared__ float Bl[32 * 33];                 // B[k][n] = W[nbase+n][kc+k], padded
    const float* G = ws + WS_G;
    v8f acc = {0.f, 0.f, 0.f, 0.f, 0.f, 0.f, 0.f, 0.f};

    const int mrow  = mt * 16 + (lane & 15);
    const int ncol  = nt * 16 + (lane & 15);
    const int khalf = (lane >> 4) * 2;            // lanes 16-31 carry K+2,K+3

    for (int kc = 0; kc < Dd; kc += 32) {
        // stage A: 64x32 floats (512 float4, 2 per thread)
        for (int it = 0; it < 2; ++it) {
            int v4 = t + it * 256;                // 0..511
            int row = v4 >> 3, c4 = (v4 & 7) * 4;
            float4 d = *(const float4*)(G + (size_t)row * Dd + kc + c4);
            float* dst = &Al[row * 33 + c4];
            dst[0] = d.x; dst[1] = d.y; dst[2] = d.z; dst[3] = d.w;
        }
        // stage B transposed: 32 rows of W, 32 contiguous k each (1 float4/thread)
        {
            int dl = t >> 3, k4 = (t & 7) * 4;
            float4 d = *(const float4*)(Wm + (size_t)(nbase + dl) * Dd + kc + k4);
            Bl[(k4 + 0) * 33 + dl] = d.x;
            Bl[(k4 + 1) * 33 + dl] = d.y;
            Bl[(k4 + 2) * 33 + dl] = d.z;
            Bl[(k4 + 3) * 33 + dl] = d.w;
        }
        __syncthreads();
        for (int k = 0; k < 32; k += 4) {
            v2f a, bv;
            a.x  = Al[mrow * 33 + k + khalf];
            a.y  = Al[mrow * 33 + k + khalf + 1];
            bv.x = Bl[(k + khalf) * 33 + ncol];
            bv.y = Bl[(k + khalf + 1) * 33 + ncol];
            acc = __builtin_amdgcn_wmma_f32_16x16x4_f32(
                      false, a, false, bv, (short)0, acc, false, false);
        }
        __syncthreads();
    }
    // epilogue: D layout — VGPR r: M = r + 8*(lane>=16), N = lane%16
    int n = nbase + nt * 16 + (lane & 15);
    float add = wb[n] + bias[n];
    int mb = mt * 16 + ((lane >> 4) << 3);
    for (int r = 0; r < 8; ++r) {
        float v = acc[r] + add;
        out[(size_t)(mb + r) * Dd + n] = v > 0.f ? v : 0.f;
    }
}

extern "C" void kernel_launch(void* const* d_in, const int* in_sizes, int n_in,
                              void* d_out, int out_size, void* d_ws, size_t ws_size,
                              hipStream_t stream) {
    const float* feat = (const float*)d_in[0];  // (64,2048,1024)
    const float* W    = (const float*)d_in[1];  // (1024,1024)
    const float* wb   = (const float*)d_in[2];  // (1024,)
    const float* aw   = (const float*)d_in[3];  // (2048,)
    const float* ab   = (const float*)d_in[4];  // scalar
    const float* bias = (const float*)d_in[5];  // (1024,)
    float* out = (float*)d_out;                 // (64,1,1024)
    float* ws  = (float*)d_ws;

    prep_v   <<<4, 256, 0, stream>>>(W, aw, ws);
    prep_cc  <<<1, 256, 0, stream>>>(wb, aw, ab, ws);
    s0_kernel<<<Bsz, 256, 0, stream>>>(feat, ws);
    flash_kernel<<<dim3(CH, Bsz), 256, 0, stream>>>(feat, ws);
    combine_kernel<<<Bsz, 256, 0, stream>>>(ws);
    wmma_gemm<<<Dd / 32, 256, 0, stream>>>(W, wb, bias, out, ws);
}